// GNN_35089882808863
// MI455X (gfx1250) — compile-verified
//
#include <hip/hip_runtime.h>
#include <hip/hip_bf16.h>

typedef __attribute__((ext_vector_type(2))) float v2f;
typedef __attribute__((ext_vector_type(8))) float v8f;

namespace {

constexpr int N_NODES = 50000;
constexpr int DEG     = 32;
constexpr int E_EDGES = N_NODES * DEG;   // 1,600,000
constexpr int HID     = 64;

// ---------------------------------------------------------------------------
// B fragment (4x16 slice of weight W[fin x 64], row-major) for wmma f32 16x16x4.
// Lane group h (lane>=16) holds K = k0+2h (x) and k0+2h+1 (y), N = ntile*16 + (lane&15).
// Rows >= fin are zeroed (handles K padding 193->196, 257->260, 3->4).
__device__ __forceinline__ v2f bfrag(const float* __restrict__ W, int fin,
                                     int k0, int half, int col) {
  int kx = k0 + 2 * half;
  int ka = kx < fin ? kx : 0;           // clamp to avoid OOB speculation
  int kb = (kx + 1) < fin ? (kx + 1) : 0;
  float x = W[(size_t)ka * HID + col];
  float y = W[(size_t)kb * HID + col];
  v2f b;
  b.x = (kx < fin) ? x : 0.0f;
  b.y = ((kx + 1) < fin) ? y : 0.0f;
  return b;
}

// ---- A-fragment loaders (per-lane row pointers captured once per tile) ----
struct ALoadEdge0 {            // edge_attr row of 3 floats, K padded to 4
  const float* p;
  __device__ __forceinline__ v2f operator()(int /*k0*/, int half) const {
    v2f a;
    if (half == 0) { a.x = p[0]; a.y = p[1]; }
    else           { a.x = p[2]; a.y = 0.0f; }   // K=3 is pad -> 0 (avoid NaN*0)
    return a;
  }
};
struct ALoadEdge {             // concat(h[row], h[col], e) : 3 x 64 segments
  const float* ph; const float* pc; const float* pe;
  __device__ __forceinline__ v2f operator()(int k0, int half) const {
    const float* p = (k0 < 64) ? ph : ((k0 < 128) ? pc : pe);
    return *(const v2f*)(p + (k0 & 63) + 2 * half);   // 8B aligned (even offsets)
  }
};
struct ALoadRow {              // dense row-major matrix (nin), zero-padded cols
  const float* p;
  __device__ __forceinline__ v2f operator()(int k0, int half) const {
    return *(const v2f*)(p + k0 + 2 * half);
  }
};

// ---------------------------------------------------------------------------
// One wave computes a 16-row tile of  out = (relu(A@W1+b1))@W2 + b2  -> c2[4] (16x64).
// h1lds: per-wave 16x64 LDS staging for the ReLU intermediate (same-wave LDS is
// in-order, so no barrier is required).
template <typename AL>
__device__ __forceinline__ void mlp_tile(const AL& al, int fin1, int kpad1,
    const float* __restrict__ W1, const float* __restrict__ B1,
    const float* __restrict__ W2, const float* __restrict__ B2,
    float* h1lds, int half, int l15, v8f c2[4]) {
  v8f c1[4];
#pragma unroll
  for (int nt = 0; nt < 4; ++nt)
#pragma unroll
    for (int r = 0; r < 8; ++r) c1[nt][r] = 0.0f;

  for (int k0 = 0; k0 < kpad1; k0 += 4) {
    v2f a = al(k0, half);
#pragma unroll
    for (int nt = 0; nt < 4; ++nt) {
      v2f b = bfrag(W1, fin1, k0, half, nt * 16 + l15);
      c1[nt] = __builtin_amdgcn_wmma_f32_16x16x4_f32(
          false, a, false, b, (short)0, c1[nt], false, false);
    }
  }
  // bias + ReLU, stage to LDS as row-major 16x64 A-tile for the second GEMM
#pragma unroll
  for (int nt = 0; nt < 4; ++nt) {
    float bias = B1[nt * 16 + l15];
#pragma unroll
    for (int r = 0; r < 8; ++r) {
      float v = c1[nt][r] + bias;
      h1lds[(size_t)(r + 8 * half) * HID + nt * 16 + l15] = v > 0.0f ? v : 0.0f;
    }
  }
#pragma unroll
  for (int nt = 0; nt < 4; ++nt)
#pragma unroll
    for (int r = 0; r < 8; ++r) c2[nt][r] = 0.0f;

  for (int k0 = 0; k0 < HID; k0 += 4) {
    v2f a = *(const v2f*)(h1lds + (size_t)l15 * HID + k0 + 2 * half);
#pragma unroll
    for (int nt = 0; nt < 4; ++nt) {
      v2f b = bfrag(W2, HID, k0, half, nt * 16 + l15);
      c2[nt] = __builtin_amdgcn_wmma_f32_16x16x4_f32(
          false, a, false, b, (short)0, c2[nt], false, false);
    }
  }
#pragma unroll
  for (int nt = 0; nt < 4; ++nt) {
    float bias = B2[nt * 16 + l15];
#pragma unroll
    for (int r = 0; r < 8; ++r) c2[nt][r] += bias;
  }
}

// ---------------------------------------------------------------------------
__global__ void k_edge0(const float* __restrict__ ea,
                        const float* __restrict__ W1, const float* __restrict__ b1,
                        const float* __restrict__ W2, const float* __restrict__ b2,
                        float* __restrict__ e, int ntiles) {
  extern __shared__ float smem[];
  int lane = threadIdx.x & 31, wv = threadIdx.x >> 5;
  int half = lane >> 4, l15 = lane & 15;
  float* h1 = smem + (size_t)wv * 16 * HID;
  int nw = blockDim.x >> 5;
  for (int tile = blockIdx.x * nw + wv; tile < ntiles; tile += gridDim.x * nw) {
    int row0 = tile * 16;
    ALoadEdge0 al{ea + (size_t)(row0 + l15) * 3};
    v8f c2[4];
    mlp_tile(al, 3, 4, W1, b1, W2, b2, h1, half, l15, c2);
    float* ep = e + (size_t)row0 * HID;
#pragma unroll
    for (int nt = 0; nt < 4; ++nt)
#pragma unroll
      for (int r = 0; r < 8; ++r)
        ep[(size_t)(r + 8 * half) * HID + nt * 16 + l15] = c2[nt][r];
  }
}

__global__ void k_edge(const float* __restrict__ h, const int* __restrict__ eidx,
                       float* e,  // read (A seg2 + residual) then overwritten in place
                       const float* __restrict__ W1, const float* __restrict__ b1,
                       const float* __restrict__ W2, const float* __restrict__ b2,
                       int ntiles) {
  extern __shared__ float smem[];
  int lane = threadIdx.x & 31, wv = threadIdx.x >> 5;
  int half = lane >> 4, l15 = lane & 15;
  float* h1 = smem + (size_t)wv * 16 * HID;
  int nw = blockDim.x >> 5;
  for (int tile = blockIdx.x * nw + wv; tile < ntiles; tile += gridDim.x * nw) {
    int row0 = tile * 16;
    int edge = row0 + l15;
    ALoadEdge al;
    al.ph = h + (size_t)eidx[edge] * HID;            // h[row]
    al.pc = h + (size_t)eidx[E_EDGES + edge] * HID;  // h[col]
    al.pe = e + (size_t)edge * HID;                  // e
    v8f c2[4];
    mlp_tile(al, 192, 192, W1, b1, W2, b2, h1, half, l15, c2);
    // residual + in-place store (rows of this tile are only touched by this wave)
    float* ep = e + (size_t)row0 * HID;
#pragma unroll
    for (int nt = 0; nt < 4; ++nt)
#pragma unroll
      for (int r = 0; r < 8; ++r) {
        size_t idx = (size_t)(r + 8 * half) * HID + nt * 16 + l15;
        ep[idx] = c2[nt][r] + ep[idx];
      }
  }
}

__global__ void k_node(const float* __restrict__ nin, int stride, int kpad, int fin,
                       const float* __restrict__ W1, const float* __restrict__ b1,
                       const float* __restrict__ W2, const float* __restrict__ b2,
                       float* __restrict__ hout, int residual, int ntiles) {
  extern __shared__ float smem[];
  int lane = threadIdx.x & 31, wv = threadIdx.x >> 5;
  int half = lane >> 4, l15 = lane & 15;
  float* h1 = smem + (size_t)wv * 16 * HID;
  int nw = blockDim.x >> 5;
  for (int tile = blockIdx.x * nw + wv; tile < ntiles; tile += gridDim.x * nw) {
    int row0 = tile * 16;
    ALoadRow al{nin + (size_t)(row0 + l15) * stride};
    v8f c2[4];
    mlp_tile(al, fin, kpad, W1, b1, W2, b2, h1, half, l15, c2);
#pragma unroll
    for (int nt = 0; nt < 4; ++nt)
#pragma unroll
      for (int r = 0; r < 8; ++r) {
        int m = r + 8 * half, n = nt * 16 + l15;
        float v = c2[nt][r];
        if (residual)  // h_old lives in nin cols [0,64)
          v += nin[(size_t)(row0 + m) * stride + n];
        hout[(size_t)(row0 + m) * HID + n] = v;
      }
  }
}

// Aggregation: edges of node v are exactly [32v, 32v+32) (row = repeat(arange(N),32)),
// cnt == 32 for every node.  Builds the zero-padded node-MLP input matrix.
__global__ void k_agg(const float* __restrict__ e, const float* __restrict__ h,
                      const float* __restrict__ u, const int* __restrict__ batch,
                      float* __restrict__ nin, int stride, int n) {
  int t = blockIdx.x * blockDim.x + threadIdx.x;
  int v = t >> 6, c = t & 63;
  if (v >= n) return;
  const float* ep = e + (size_t)v * DEG * HID + c;
  float s = 0.0f, mx = -3.402823466e38f;
#pragma unroll 4
  for (int k = 0; k < DEG; ++k) {
    float x = ep[(size_t)k * HID];
    s += x; mx = fmaxf(mx, x);
  }
  float* np = nin + (size_t)v * stride;
  int base = 0;
  if (h != nullptr) { np[c] = h[(size_t)v * HID + c]; base = HID; }
  np[base + c]           = s;
  np[base + HID + c]     = mx;
  np[base + 2 * HID + c] = s * (1.0f / DEG);
  int ucol = base + 3 * HID;
  if (c == 0) np[ucol] = u[batch[v]];
  if (c < 3)  np[ucol + 1 + c] = 0.0f;   // K padding to multiple of 4
}

// Global pooling over 4 graphs (12500 contiguous nodes each) + tiny out-MLP.
__global__ void k_out(const float* __restrict__ h, const float* __restrict__ u,
                      const float* __restrict__ W0, const float* __restrict__ b0,
                      const float* __restrict__ W1, const float* __restrict__ b1,
                      const float* __restrict__ W2, const float* __restrict__ b2,
                      const float* __restrict__ W3, const float* __restrict__ b3,
                      float* __restrict__ out, int n) {
  __shared__ float ob[4 * 193];
  __shared__ float t1[4 * HID];
  __shared__ float t2[4 * HID];
  int t = threadIdx.x, g = t >> 6, c = t & 63;
  int np = n / 4;
  const float* hp = h + (size_t)g * np * HID + c;
  float s = 0.0f, mx = -3.402823466e38f;
  for (int v = 0; v < np; ++v) {
    float x = hp[(size_t)v * HID];
    s += x; mx = fmaxf(mx, x);
  }
  ob[g * 193 + c]            = s;
  ob[g * 193 + HID + c]      = s / (float)np;
  ob[g * 193 + 2 * HID + c]  = mx;
  if (c == 0) ob[g * 193 + 192] = u[g];
  __syncthreads();
  { const float* o = ob + g * 193; float a = b0[c];
    for (int k = 0; k < 193; ++k) a += o[k] * W0[k * HID + c];
    t1[g * HID + c] = fmaxf(a, 0.0f); }
  __syncthreads();
  { const float* o = t1 + g * HID; float a = b1[c];
    for (int k = 0; k < HID; ++k) a += o[k] * W1[k * HID + c];
    t2[g * HID + c] = fmaxf(a, 0.0f); }
  __syncthreads();
  { const float* o = t2 + g * HID; float a = b2[c];
    for (int k = 0; k < HID; ++k) a += o[k] * W2[k * HID + c];
    t1[g * HID + c] = fmaxf(a, 0.0f); }
  __syncthreads();
  if (c < 2) { const float* o = t1 + g * HID; float a = b3[c];
    for (int k = 0; k < HID; ++k) a += o[k] * W3[k * 2 + c];
    out[g * 2 + c] = a; }
}

}  // namespace

// d_in: 0=x(unused) 1=edge_attr 2=u 3..34=params(32 leaves) 35=edge_index 36=batch
// ws layout: e[E*64] | h[N*64] | nin[N*260]   (~475 MB total)
extern "C" void kernel_launch(void* const* d_in, const int* in_sizes, int n_in,
                              void* d_out, int out_size, void* d_ws, size_t ws_size,
                              hipStream_t stream) {
  (void)in_sizes; (void)n_in; (void)out_size; (void)ws_size;
  const float* edge_attr = (const float*)d_in[1];
  const float* u         = (const float*)d_in[2];
  const float* P[32];
  for (int i = 0; i < 32; ++i) P[i] = (const float*)d_in[3 + i];
  const int* eidx  = (const int*)d_in[35];
  const int* batch = (const int*)d_in[36];

  float* e   = (float*)d_ws;
  float* h   = e + (size_t)E_EDGES * HID;
  float* nin = h + (size_t)N_NODES * HID;

  const int ntE = E_EDGES / 16;   // 100000
  const int ntN = N_NODES / 16;   // 3125
  dim3 blk(128);                  // 4 waves
  size_t smem = (size_t)4 * 16 * HID * sizeof(float);   // 16 KB h1 staging
  const int aggBlocks = (N_NODES * HID) / 256;

  // layer 0 (only_positions)
  k_edge0<<<1250, blk, smem, stream>>>(edge_attr, P[0], P[1], P[2], P[3], e, ntE);
  k_agg<<<aggBlocks, 256, 0, stream>>>(e, nullptr, u, batch, nin, 196, N_NODES);
  k_node<<<200, blk, smem, stream>>>(nin, 196, 196, 193, P[4], P[5], P[6], P[7],
                                     h, 0, ntN);
  // layers 1..2 with residuals
  for (int L = 1; L <= 2; ++L) {
    const float* const* q = P + 8 * L;
    k_edge<<<1250, blk, smem, stream>>>(h, eidx, e, q[0], q[1], q[2], q[3], ntE);
    k_agg<<<aggBlocks, 256, 0, stream>>>(e, h, u, batch, nin, 260, N_NODES);
    k_node<<<200, blk, smem, stream>>>(nin, 260, 260, 257, q[4], q[5], q[6], q[7],
                                       h, 1, ntN);
  }
  k_out<<<1, 256, 0, stream>>>(h, u, P[24], P[25], P[26], P[27], P[28], P[29],
                               P[30], P[31], (float*)d_out, N_NODES);
}